// TNEP_16947940950580
// MI455X (gfx1250) — compile-verified
//
#include <hip/hip_runtime.h>
#include <math.h>

typedef float v2f __attribute__((ext_vector_type(2)));
typedef float v8f __attribute__((ext_vector_type(8)));

#define A_N 4096
#define M_N 64
#define Q_N 64
#define H_N 64
#define T_N 4

// ---------------------------------------------------------------------------
// Masked per-type GEMM tile: D[16 rows x 64 cols] += A[16x64] x W[Z[row]]
// A rows come from LDS; W is [T][Q][H] in global (64KB total -> cache-resident).
// TRANS=false : B[k][col] = W[t][k][col]   (forward:  sum_q  desc * W0)
// TRANS=true  : B[k][col] = W[t][col][k]   (backward: sum_h  de_da * W0^T)
// Accumulation over the 4 types with per-row masks gives the per-atom-type
// result exactly (each row matches exactly one type).
// WMMA f32 16x16x4 layouts (wave32):
//   A: lane l (both halves) holds row M=(l&15); VGPR0/1 = K = 4*k4+2*half +0/1
//   B: lane l holds col N=(l&15);               VGPR0/1 = K = 4*k4+2*half +0/1
//   C/D: VGPR j, lane l -> row M = j + 8*(l>>4), col N = (l&15) + 16*ct
// ---------------------------------------------------------------------------
template <bool TRANS>
__device__ inline void wmma_gemm16x64(const float (&sA)[16][64],
                                      const int* sZ,
                                      const float* __restrict__ W,
                                      int lane, v8f acc[4]) {
    const int half = lane >> 4;
    const int l15  = lane & 15;
#pragma unroll
    for (int ct = 0; ct < 4; ++ct) {
        v8f c = acc[ct];
        const int col = ct * 16 + l15;
        for (int t = 0; t < T_N; ++t) {
            const float rmask = (sZ[l15] == t) ? 1.0f : 0.0f;
            const float* Wt = W + t * Q_N * H_N;
            for (int k4 = 0; k4 < 16; ++k4) {
                const int k = k4 * 4 + 2 * half;
                v2f a, b;
                a[0] = sA[l15][k]     * rmask;
                a[1] = sA[l15][k + 1] * rmask;
                if (TRANS) {
                    b[0] = Wt[col * H_N + k];
                    b[1] = Wt[col * H_N + k + 1];
                } else {
                    b[0] = Wt[k * H_N + col];
                    b[1] = Wt[(k + 1) * H_N + col];
                }
                c = __builtin_amdgcn_wmma_f32_16x16x4_f32(
                        false, a, false, b, (short)0, c, false, false);
            }
        }
        acc[ct] = c;
    }
}

// ---------------------------------------------------------------------------
// Kernel 1: per-16-atom-tile MLPs (one wave per tile).
//   h      = tanh(q @ W0[Z] + b0[Z]) * amask
//   de_da  = (1 - h^2) * W1[Z]
//   de_dq  = de_da @ W0[Z]^T            -> written to ws (used by kernel 2)
//   h_pol  = tanh(q @ W0p[Z] + b0p[Z]) * amask
//   F_pol  = (sum_h h_pol*W1p[Z] + b1p) * amask  -> atomic into acc[9]
// ---------------------------------------------------------------------------
__global__ __launch_bounds__(32) void mlp_kernel(
    const float* __restrict__ desc, const int* __restrict__ Z,
    const float* __restrict__ amask,
    const float* __restrict__ W0, const float* __restrict__ b0,
    const float* __restrict__ W1,
    const float* __restrict__ W0p, const float* __restrict__ b0p,
    const float* __restrict__ W1p, const float* __restrict__ b1p,
    float* __restrict__ dedq, float* __restrict__ acc_out) {
    __shared__ float sA[16][64];   // descriptor tile (A operand)
    __shared__ float sS[16][64];   // staging: de_da, then pol products
    __shared__ int   sZ[16];
    __shared__ float sAm[16];

    const int lane = threadIdx.x;
    const int half = lane >> 4;
    const int l15  = lane & 15;
    const int n0   = blockIdx.x * 16;

    if (lane < 16) {
        sZ[lane]  = Z[n0 + lane];
        sAm[lane] = amask[n0 + lane];
    }
    for (int i = lane; i < 16 * 64; i += 32)
        sA[i >> 6][i & 63] = desc[(size_t)(n0 + (i >> 6)) * Q_N + (i & 63)];
    __syncthreads();

    // ---- forward main MLP ----
    v8f acc[4] = {};
    wmma_gemm16x64<false>(sA, sZ, W0, lane, acc);
#pragma unroll
    for (int ct = 0; ct < 4; ++ct) {
#pragma unroll
        for (int j = 0; j < 8; ++j) {
            const int r = j + 8 * half;
            const int col = ct * 16 + l15;
            const int zr = sZ[r];
            float h = tanhf(acc[ct][j] + b0[zr * H_N + col]) * sAm[r];
            // de_da (note: dtanh computed from MASKED h, matching reference)
            sS[r][col] = (1.0f - h * h) * W1[zr * H_N + col];
        }
    }
    __syncthreads();

    // ---- backward: de_dq = de_da @ W0^T ----
    v8f accB[4] = {};
    wmma_gemm16x64<true>(sS, sZ, W0, lane, accB);
#pragma unroll
    for (int ct = 0; ct < 4; ++ct) {
#pragma unroll
        for (int j = 0; j < 8; ++j) {
            const int r = j + 8 * half;
            dedq[(size_t)(n0 + r) * Q_N + ct * 16 + l15] = accB[ct][j];
        }
    }
    __syncthreads();

    // ---- pol MLP ----
    v8f accP[4] = {};
    wmma_gemm16x64<false>(sA, sZ, W0p, lane, accP);
#pragma unroll
    for (int ct = 0; ct < 4; ++ct) {
#pragma unroll
        for (int j = 0; j < 8; ++j) {
            const int r = j + 8 * half;
            const int col = ct * 16 + l15;
            const int zr = sZ[r];
            float hp = tanhf(accP[ct][j] + b0p[zr * H_N + col]) * sAm[r];
            sS[r][col] = hp * W1p[zr * H_N + col];
        }
    }
    __syncthreads();

    float F = 0.0f;
    if (lane < 16) {
        float s = 0.0f;
        for (int c = 0; c < 64; ++c) s += sS[lane][c];
        F = (s + b1p[0]) * sAm[lane];
    }
#pragma unroll
    for (int o = 16; o >= 1; o >>= 1) F += __shfl_xor(F, o, 32);
    if (lane == 0) atomicAdd(acc_out + 9, F);
}

// ---------------------------------------------------------------------------
// Kernel 2: bandwidth stage. One wave per atom n; lane owns q in {l, l+32}.
//   f[b]        = (sum_q de_dq[q]*grad[n,m,b,q]) * nmask
//   S[a][b]    += dr[a]*f[b]*nmask   (pm = -S)
// Per-lane 3x3 accumulators; single 9-value shuffle reduce per wave; atomics.
// ---------------------------------------------------------------------------
__global__ __launch_bounds__(256) void forces_kernel(
    const float* __restrict__ grad, const int* __restrict__ gidx,
    const float* __restrict__ pos, const float* __restrict__ box,
    const float* __restrict__ nmask, const float* __restrict__ dedq,
    float* __restrict__ acc_out) {
    const int lane = threadIdx.x & 31;
    const int wave = threadIdx.x >> 5;
    const int n = blockIdx.x * 8 + wave;
    if (n >= A_N) return;

    // general 3x3 box inverse (inv = adj^T / det)
    float B[9];
#pragma unroll
    for (int i = 0; i < 9; ++i) B[i] = box[i];
    const float det = B[0] * (B[4] * B[8] - B[5] * B[7]) -
                      B[1] * (B[3] * B[8] - B[5] * B[6]) +
                      B[2] * (B[3] * B[7] - B[4] * B[6]);
    const float id = 1.0f / det;
    float Bi[9];
    Bi[0] = (B[4] * B[8] - B[5] * B[7]) * id;
    Bi[1] = (B[2] * B[7] - B[1] * B[8]) * id;
    Bi[2] = (B[1] * B[5] - B[2] * B[4]) * id;
    Bi[3] = (B[5] * B[6] - B[3] * B[8]) * id;
    Bi[4] = (B[0] * B[8] - B[2] * B[6]) * id;
    Bi[5] = (B[2] * B[3] - B[0] * B[5]) * id;
    Bi[6] = (B[3] * B[7] - B[4] * B[6]) * id;
    Bi[7] = (B[1] * B[6] - B[0] * B[7]) * id;
    Bi[8] = (B[0] * B[4] - B[1] * B[3]) * id;

    const float d0 = dedq[(size_t)n * Q_N + lane];
    const float d1 = dedq[(size_t)n * Q_N + 32 + lane];
    const float pix = pos[n * 3 + 0], piy = pos[n * 3 + 1], piz = pos[n * 3 + 2];

    float S[9];
#pragma unroll
    for (int i = 0; i < 9; ++i) S[i] = 0.0f;

    const float* gb = grad + (size_t)n * M_N * 3 * Q_N;
    for (int m = 0; m < M_N; ++m) {
        const float* g = gb + (size_t)m * 3 * Q_N;
        if (m + 1 < M_N)  // prefetch next 768B row (lanes cover all cachelines)
            __builtin_prefetch(g + 3 * Q_N + lane * 6, 0, 0);

        const int jn = gidx[n * M_N + m];
        const float nm = nmask[n * M_N + m];
        float dx = pos[jn * 3 + 0] - pix;
        float dy = pos[jn * 3 + 1] - piy;
        float dz = pos[jn * 3 + 2] - piz;
        // minimal image: frac = d @ Binv ; frac -= round(frac) ; dr = frac @ B
        float fx = dx * Bi[0] + dy * Bi[3] + dz * Bi[6];
        float fy = dx * Bi[1] + dy * Bi[4] + dz * Bi[7];
        float fz = dx * Bi[2] + dy * Bi[5] + dz * Bi[8];
        fx -= rintf(fx); fy -= rintf(fy); fz -= rintf(fz);
        const float rx = (fx * B[0] + fy * B[3] + fz * B[6]) * nm;
        const float ry = (fx * B[1] + fy * B[4] + fz * B[7]) * nm;
        const float rz = (fx * B[2] + fy * B[5] + fz * B[8]) * nm;

        const float f0 = (d0 * g[lane]       + d1 * g[32 + lane])  * nm;
        const float f1 = (d0 * g[64 + lane]  + d1 * g[96 + lane])  * nm;
        const float f2 = (d0 * g[128 + lane] + d1 * g[160 + lane]) * nm;

        S[0] += rx * f0; S[1] += rx * f1; S[2] += rx * f2;
        S[3] += ry * f0; S[4] += ry * f1; S[5] += ry * f2;
        S[6] += rz * f0; S[7] += rz * f1; S[8] += rz * f2;
    }

#pragma unroll
    for (int i = 0; i < 9; ++i) {
#pragma unroll
        for (int o = 16; o >= 1; o >>= 1) S[i] += __shfl_xor(S[i], o, 32);
    }
    if (lane == 0) {
#pragma unroll
        for (int i = 0; i < 9; ++i) atomicAdd(acc_out + i, S[i]);
    }
}

__global__ void init_kernel(float* __restrict__ acc) {
    if (threadIdx.x < 10) acc[threadIdx.x] = 0.0f;
}

__global__ void finalize_kernel(const float* __restrict__ acc,
                                float* __restrict__ out) {
    if (threadIdx.x == 0) {
        const float ss = acc[9];
        out[0] = -acc[0] + ss;  // pm[0,0] + scalar_sum
        out[1] = -acc[4] + ss;  // pm[1,1] + scalar_sum
        out[2] = -acc[8] + ss;  // pm[2,2] + scalar_sum
        out[3] = -acc[1];       // pm[0,1]
        out[4] = -acc[5];       // pm[1,2]
        out[5] = -acc[6];       // pm[2,0]
    }
}

extern "C" void kernel_launch(void* const* d_in, const int* in_sizes, int n_in,
                              void* d_out, int out_size, void* d_ws,
                              size_t ws_size, hipStream_t stream) {
    const float* desc  = (const float*)d_in[0];
    const float* grad  = (const float*)d_in[1];
    const int*   gidx  = (const int*)d_in[2];
    const float* pos   = (const float*)d_in[3];
    const int*   Z     = (const int*)d_in[4];
    const float* box   = (const float*)d_in[5];
    const float* amask = (const float*)d_in[6];
    const float* nmask = (const float*)d_in[7];
    const float* W0    = (const float*)d_in[8];
    const float* b0    = (const float*)d_in[9];
    const float* W1    = (const float*)d_in[10];
    /* d_in[11] = b1 (unused by reference) */
    const float* W0p   = (const float*)d_in[12];
    const float* b0p   = (const float*)d_in[13];
    const float* W1p   = (const float*)d_in[14];
    const float* b1p   = (const float*)d_in[15];

    float* dedq = (float*)d_ws;                // A*Q floats = 1 MB
    float* acc  = dedq + (size_t)A_N * Q_N;    // 10 floats: S[9] + scalar_sum
    float* out  = (float*)d_out;

    init_kernel<<<1, 32, 0, stream>>>(acc);
    mlp_kernel<<<A_N / 16, 32, 0, stream>>>(desc, Z, amask, W0, b0, W1, W0p,
                                            b0p, W1p, b1p, dedq, acc);
    forces_kernel<<<A_N / 8, 256, 0, stream>>>(grad, gidx, pos, box, nmask,
                                               dedq, acc);
    finalize_kernel<<<1, 32, 0, stream>>>(acc, out);
}